// VectorQuantizer_12455405159140
// MI455X (gfx1250) — compile-verified
//
#include <hip/hip_runtime.h>
#include <hip/hip_bf16.h>

#define N_TOK   65536
#define N_E     256
#define E_DIM   32
#define SK_EPS  0.003f
#define SK_IT   100
#define USTEP_BLOCKS 512
#define USTEP_CHUNK  128   // 512*128 = 65536

typedef float v2f __attribute__((ext_vector_type(2)));
typedef float v8f __attribute__((ext_vector_type(8)));

// ---- ordered-float <-> uint encoding (monotone), for atomic min/max ----
__device__ __forceinline__ unsigned fenc(float f) {
    unsigned u = __float_as_uint(f);
    return (u & 0x80000000u) ? ~u : (u | 0x80000000u);
}
__device__ __forceinline__ float fdec(unsigned k) {
    return __uint_as_float((k & 0x80000000u) ? (k & 0x7FFFFFFFu) : ~k);
}

// ---- K0a: codebook norms + init min/max scalars ----
__global__ __launch_bounds__(256) void k_cnorm_init(const float* __restrict__ cb,
                                                    float* __restrict__ cnorm,
                                                    unsigned* __restrict__ mm) {
    int t = threadIdx.x;
    float s = 0.f;
    const float* r = cb + t * E_DIM;
    #pragma unroll
    for (int j = 0; j < E_DIM; ++j) s += r[j] * r[j];
    cnorm[t] = s;
    if (t == 0) { mm[0] = 0xFFFFFFFFu; mm[1] = 0u; }  // min-enc, max-enc
}

// ---- K0b: token norms ----
__global__ __launch_bounds__(256) void k_xnorm(const float* __restrict__ x,
                                               float* __restrict__ xnorm) {
    int n = blockIdx.x * 256 + threadIdx.x;
    const float4* r = (const float4*)(x + (size_t)n * E_DIM);
    float s = 0.f;
    #pragma unroll
    for (int j = 0; j < E_DIM / 4; ++j) {
        float4 q = r[j];
        s += q.x * q.x + q.y * q.y + q.z * q.z + q.w * q.w;
    }
    xnorm[n] = s;
}

// ---- K1: WMMA f32 GEMM -> D[n][k], plus global min/max of d ----
// one wave per 16x16 tile; 8 x V_WMMA_F32_16X16X4_F32 over K=32
__global__ __launch_bounds__(256) void k_dist(const float* __restrict__ x,
                                              const float* __restrict__ cb,
                                              const float* __restrict__ xnorm,
                                              const float* __restrict__ cnorm,
                                              float* __restrict__ D,
                                              unsigned* __restrict__ mm) {
    int wave = threadIdx.x >> 5;
    int lane = threadIdx.x & 31;
    int tile = blockIdx.x * 8 + wave;   // 65536 tiles
    int rt = tile >> 4;                 // 0..4095 row tile
    int ct = tile & 15;                 // 0..15  col tile
    int m  = lane & 15;
    int kh = lane >> 4;                 // 0/1 : K-half select

    // A: 16x4 f32 layout -> lane m rows, VGPR0/1 = K pair (kh*2, kh*2+1)
    const float* xr = x  + ((size_t)(rt * 16 + m)) * E_DIM + kh * 2;
    // B: 4x16 f32 layout mirrors A; B[k][n] = codebook[n][k]
    const float* br = cb + ((size_t)(ct * 16 + m)) * E_DIM + kh * 2;

    v8f acc = {};
    #pragma unroll
    for (int kk = 0; kk < 8; ++kk) {
        v2f a, b;
        a.x = xr[kk * 4]; a.y = xr[kk * 4 + 1];
        b.x = br[kk * 4]; b.y = br[kk * 4 + 1];
        acc = __builtin_amdgcn_wmma_f32_16x16x4_f32(false, a, false, b,
                                                    (short)0, acc, false, false);
    }

    float cn = cnorm[ct * 16 + m];
    float tmin = 3.4e38f, tmax = -3.4e38f;
    #pragma unroll
    for (int r = 0; r < 8; ++r) {
        int row = rt * 16 + r + kh * 8;      // C layout: lanes16-31 hold M+8
        float d = xnorm[row] + cn - 2.0f * acc[r];
        D[(size_t)row * N_E + ct * 16 + m] = d;
        tmin = fminf(tmin, d);
        tmax = fmaxf(tmax, d);
    }
    #pragma unroll
    for (int off = 16; off; off >>= 1) {
        tmin = fminf(tmin, __shfl_xor(tmin, off));
        tmax = fmaxf(tmax, __shfl_xor(tmax, off));
    }
    if (lane == 0) {
        atomicMin(&mm[0], fenc(tmin));
        atomicMax(&mm[1], fenc(tmax));
    }
}

// ---- K2: A[n][k] = exp(-(d - rowmin_n)/(amplitude*eps)); init v=1 ----
// (middle cancels; per-token rescale is absorbed into v — stabilized Sinkhorn)
__global__ __launch_bounds__(256) void k_buildA(float* __restrict__ A,
                                                const unsigned* __restrict__ mm,
                                                float* __restrict__ v) {
    int wave = threadIdx.x >> 5, lane = threadIdx.x & 31;
    int n = blockIdx.x * 8 + wave;
    float dmin = fdec(mm[0]), dmax = fdec(mm[1]);
    float middle = 0.5f * (dmax + dmin);
    float amplitude = dmax - middle + 1e-5f;
    float inv = 1.0f / (amplitude * SK_EPS);

    float dv[8];
    float rmin = 3.4e38f;
    #pragma unroll
    for (int j = 0; j < 8; ++j) {
        dv[j] = A[(size_t)n * N_E + lane + 32 * j];
        rmin = fminf(rmin, dv[j]);
    }
    #pragma unroll
    for (int off = 16; off; off >>= 1) rmin = fminf(rmin, __shfl_xor(rmin, off));
    #pragma unroll
    for (int j = 0; j < 8; ++j)
        A[(size_t)n * N_E + lane + 32 * j] = __expf(-(dv[j] - rmin) * inv);
    if (lane == 0) v[n] = 1.0f;
}

// ---- K3: u-step partials: partial[b][k] = sum_{n in chunk} A[n][k]*v[n] ----
__global__ __launch_bounds__(256) void k_ustep(const float* __restrict__ A,
                                               const float* __restrict__ v,
                                               float* __restrict__ uPartial) {
    int t = threadIdx.x;
    int n0 = blockIdx.x * USTEP_CHUNK;
    float acc = 0.f;
    for (int i = 0; i < USTEP_CHUNK; ++i) {
        int n = n0 + i;
        if (i + 8 < USTEP_CHUNK)
            __builtin_prefetch(&A[(size_t)(n + 8) * N_E + t], 0, 0);
        acc += A[(size_t)n * N_E + t] * v[n];
    }
    uPartial[blockIdx.x * N_E + t] = acc;
}

// ---- K4: u[k] = 1/(K * sum_p partial[p][k]) (fixed-order, deterministic) ----
__global__ __launch_bounds__(256) void k_ufin(const float* __restrict__ uPartial,
                                              float* __restrict__ u) {
    int t = threadIdx.x;
    float s = 0.f;
    for (int p = 0; p < USTEP_BLOCKS; ++p) s += uPartial[p * N_E + t];
    u[t] = 1.0f / (256.0f * s);
}

// ---- K5: v[n] = 1/(N * sum_k A[n][k]*u[k]); wave per token ----
__global__ __launch_bounds__(256) void k_vstep(const float* __restrict__ A,
                                               const float* __restrict__ u,
                                               float* __restrict__ v) {
    __shared__ float us[N_E];
    us[threadIdx.x] = u[threadIdx.x];
    __syncthreads();
    int wave = threadIdx.x >> 5, lane = threadIdx.x & 31;
    int n = blockIdx.x * 8 + wave;
    float s = 0.f;
    #pragma unroll
    for (int j = 0; j < 8; ++j) {
        int k = lane + 32 * j;
        s += A[(size_t)n * N_E + k] * us[k];
    }
    #pragma unroll
    for (int off = 16; off; off >>= 1) s += __shfl_xor(s, off);
    if (lane == 0) v[n] = 1.0f / (65536.0f * s);
}

// ---- K6: argmax_k u_k*A[n][k] -> index, x_q, per-block loss partial ----
__global__ __launch_bounds__(256) void k_final(const float* __restrict__ A,
                                               const float* __restrict__ u,
                                               const float* __restrict__ x,
                                               const float* __restrict__ cb,
                                               const int* __restrict__ use_sk,
                                               float* __restrict__ out_xq,
                                               float* __restrict__ out_idx,
                                               float* __restrict__ lossPartial) {
    __shared__ float us[N_E];
    __shared__ float wsum[8];
    int sk = use_sk[0];
    us[threadIdx.x] = sk ? u[threadIdx.x] : 1.0f;   // use_sk==0 -> argmin d == argmax A
    __syncthreads();
    int wave = threadIdx.x >> 5, lane = threadIdx.x & 31;
    int n = blockIdx.x * 8 + wave;

    float bv = -3.4e38f; int bi = 0;
    #pragma unroll
    for (int j = 0; j < 8; ++j) {
        int k = lane + 32 * j;
        float q = A[(size_t)n * N_E + k] * us[k];
        if (q > bv) { bv = q; bi = k; }             // within-lane k ascends: first max kept
    }
    #pragma unroll
    for (int off = 16; off; off >>= 1) {
        float ov = __shfl_xor(bv, off);
        int   oi = __shfl_xor(bi, off);
        if (ov > bv || (ov == bv && oi < bi)) { bv = ov; bi = oi; }
    }
    // all lanes now agree on bi; lane j writes element j of token n
    float c  = cb[(size_t)bi * E_DIM + lane];
    float xv = x[(size_t)n * E_DIM + lane];
    out_xq[(size_t)n * E_DIM + lane] = xv + (c - xv);  // straight-through
    float diff = c - xv;
    float ls = diff * diff;
    #pragma unroll
    for (int off = 16; off; off >>= 1) ls += __shfl_xor(ls, off);
    if (lane == 0) { out_idx[n] = (float)bi; wsum[wave] = ls; }
    __syncthreads();
    if (threadIdx.x == 0) {
        float s = 0.f;
        #pragma unroll
        for (int w = 0; w < 8; ++w) s += wsum[w];
        lossPartial[blockIdx.x] = s;
    }
}

// ---- K7: loss = 1.25 * sum / (N*E_DIM), fixed-order reduce ----
__global__ __launch_bounds__(256) void k_loss(const float* __restrict__ lossPartial,
                                              float* __restrict__ out_loss) {
    __shared__ float sm[256];
    int t = threadIdx.x;
    float s = 0.f;
    for (int i = 0; i < 32; ++i) s += lossPartial[t + 256 * i];  // 8192 partials
    sm[t] = s;
    __syncthreads();
    for (int off = 128; off; off >>= 1) {
        if (t < off) sm[t] += sm[t + off];
        __syncthreads();
    }
    if (t == 0) out_loss[0] = 1.25f * sm[0] / (float)(N_TOK * E_DIM);
}

extern "C" void kernel_launch(void* const* d_in, const int* in_sizes, int n_in,
                              void* d_out, int out_size, void* d_ws, size_t ws_size,
                              hipStream_t stream) {
    (void)in_sizes; (void)n_in; (void)out_size; (void)ws_size;
    const float* x      = (const float*)d_in[0];
    const float* cb     = (const float*)d_in[1];
    const int*   use_sk = (const int*)d_in[2];
    float* out = (float*)d_out;
    float* out_xq   = out;                  // 65536*32
    float* out_loss = out + (size_t)N_TOK * E_DIM;          // 1
    float* out_idx  = out_loss + 1;                         // 65536 (as float)

    char* ws = (char*)d_ws;
    float* A        = (float*)ws;                           // 64 MB: D then exp-matrix
    float* uPartial = (float*)(ws + (size_t)N_TOK * N_E * sizeof(float));
    float* u        = uPartial + USTEP_BLOCKS * N_E;
    float* v        = u + N_E;
    float* lossP    = v + N_TOK;
    float* xnorm    = lossP + 8192;
    float* cnorm    = xnorm + N_TOK;
    unsigned* mm    = (unsigned*)(cnorm + N_E);

    k_cnorm_init<<<1, 256, 0, stream>>>(cb, cnorm, mm);
    k_xnorm<<<N_TOK / 256, 256, 0, stream>>>(x, xnorm);
    k_dist<<<8192, 256, 0, stream>>>(x, cb, xnorm, cnorm, A, mm);
    k_buildA<<<8192, 256, 0, stream>>>(A, mm, v);

    for (int it = 0; it < SK_IT; ++it) {
        k_ustep<<<USTEP_BLOCKS, 256, 0, stream>>>(A, v, uPartial);
        k_ufin<<<1, 256, 0, stream>>>(uPartial, u);
        if (it != SK_IT - 1)                 // final v-step doesn't affect argmax
            k_vstep<<<8192, 256, 0, stream>>>(A, u, v);
    }

    k_final<<<8192, 256, 0, stream>>>(A, u, x, cb, use_sk, out_xq, out_idx, lossP);
    k_loss<<<1, 256, 0, stream>>>(lossP, out_loss);
}